// Sampler_75539884802984
// MI455X (gfx1250) — compile-verified
//
#include <hip/hip_runtime.h>
#include <stdint.h>

#ifndef __has_builtin
#define __has_builtin(x) 0
#endif

// Sampler for (B=128, V=128000), MAX_TOP_K=64.
// One workgroup (256 threads = 8 wave32 waves) per row.
// Radix-select top-64 via 4x 8-bit LDS histogram rounds, fused with the
// max / sum-exp / logprob-write streaming passes (b128 vector loads).
// Pass D demonstrates the CDNA5 Tensor Data Mover: wave0 DMAs row chunks
// into LDS (tensor_load_to_lds + s_wait_tensorcnt), all waves histogram
// from LDS. Whole logits tensor (65.5MB) is L2-resident (192MB), so the
// HBM floor is ~131MB (one read + logprob write) ~= 6us at 23.3 TB/s.

#define TPB   256
#define TOPK  64
#define CAP   256    // candidate buffer (== TPB for the bitonic sort)
#define CHUNK 8192   // floats per TDM stage (32 KB)

typedef uint32_t v4u_t __attribute__((ext_vector_type(4)));
typedef int      v8i_t __attribute__((ext_vector_type(8)));
typedef int      v4i_t __attribute__((ext_vector_type(4)));

__device__ __forceinline__ uint32_t f2ord(float f) {
  uint32_t u = __float_as_uint(f);
  return (u & 0x80000000u) ? ~u : (u | 0x80000000u);   // monotone: bigger float -> bigger key
}
__device__ __forceinline__ float ord2f(uint32_t k) {
  uint32_t u = (k & 0x80000000u) ? (k ^ 0x80000000u) : ~k;
  return __uint_as_float(u);
}

// thread 0 only: pick the bin holding the 'remaining'-th largest, update prefix.
__device__ __forceinline__ void radix_resolve(uint32_t* hist, uint32_t* s_prefix,
                                              uint32_t* s_remaining) {
  uint32_t rem = *s_remaining, cum = 0;
  int b = 255;
  for (; b > 0; --b) { uint32_t h = hist[b]; if (cum + h >= rem) break; cum += h; }
  *s_remaining = rem - cum;
  *s_prefix = (*s_prefix << 8) | (uint32_t)b;
}

// 1-D TDM copy: nelem f32 from gsrc -> LDS byte offset lds_byte_off (nelem <= 65535).
__device__ __forceinline__ void tdm_load_row_chunk(uint32_t lds_byte_off,
                                                   const float* gsrc,
                                                   uint32_t nelem) {
#if __has_builtin(__builtin_amdgcn_tensor_load_to_lds)
  uint64_t ga = (uint64_t)(uintptr_t)gsrc;
  v4u_t g0;
  g0.x = 1u;                                             // count=1, user mode
  g0.y = lds_byte_off;                                   // lds_addr (bytes)
  g0.z = (uint32_t)ga;                                   // global_addr[31:0]
  g0.w = (uint32_t)((ga >> 32) & 0x01FFFFFFu) | 0x80000000u; // addr[56:32] | type=2
  v8i_t g1;
  g1[0] = (int)0x00020000u;                              // data_size=2 -> 4 bytes
  g1[1] = (int)((nelem & 0xFFFFu) << 16);                // tensor_dim0[15:0]
  g1[2] = (int)((nelem >> 16) | (1u << 16));             // tensor_dim0[31:16] | tensor_dim1=1
  g1[3] = (int)((nelem & 0xFFFFu) << 16);                // tile_dim0 = nelem (1-D tile)
  g1[4] = 0;                                             // tile_dim1=0, tile_dim2=0
  g1[5] = (int)nelem;                                    // tensor_dim0_stride[31:0]
  g1[6] = 0;
  g1[7] = 0;
  v4i_t gz = {0, 0, 0, 0};
#if defined(__clang_major__) && (__clang_major__ >= 23)
  v8i_t gz8 = {0, 0, 0, 0, 0, 0, 0, 0};
  __builtin_amdgcn_tensor_load_to_lds(g0, g1, gz, gz, gz8, 0);
#else
  __builtin_amdgcn_tensor_load_to_lds(g0, g1, gz, gz, 0);
#endif
#endif
}

__global__ __launch_bounds__(TPB) void sampler_topk_kernel(
    const float* __restrict__ logits,
    const float* __restrict__ temps,
    const int*   __restrict__ topks,
    const float* __restrict__ topps,
    const float* __restrict__ minps,
    const float* __restrict__ us,
    float* __restrict__ out, int B, int V)
{
  const int row = blockIdx.x;
  const int tid = threadIdx.x;
  const float* __restrict__ lrow = logits + (size_t)row * (size_t)V;
  float* __restrict__ lpout = out + B + (size_t)row * (size_t)V;
  int* idout = (int*)out;

  __shared__ uint32_t hist[256];
  __shared__ float    red[256];
  __shared__ unsigned long long cand[CAP];
  __shared__ float    stage[CHUNK];                       // 32 KB TDM staging
  __shared__ uint32_t s_prefix, s_remaining, s_thr, s_ccount;
  __shared__ float    s_m, s_lse;

  const float invT = 1.0f / temps[row];
  const int   V4 = V & ~3;

  // ---------------- Pass A: row max + radix round 0 (key bits [31:24]) ----
  hist[tid] = 0;
  __syncthreads();
  float lmax = -INFINITY;
  for (int i = tid * 4; i < V4; i += TPB * 4) {
    __builtin_prefetch(lrow + i + TPB * 16, 0, 3);        // global_prefetch_b8, row is re-read
    float4 x = *reinterpret_cast<const float4*>(lrow + i);
    lmax = fmaxf(fmaxf(fmaxf(x.x, x.y), fmaxf(x.z, x.w)), lmax);
    atomicAdd(&hist[f2ord(x.x) >> 24], 1u);
    atomicAdd(&hist[f2ord(x.y) >> 24], 1u);
    atomicAdd(&hist[f2ord(x.z) >> 24], 1u);
    atomicAdd(&hist[f2ord(x.w) >> 24], 1u);
  }
  for (int i = V4 + tid; i < V; i += TPB) {
    float x = lrow[i];
    lmax = fmaxf(lmax, x);
    atomicAdd(&hist[f2ord(x) >> 24], 1u);
  }
  for (int off = 16; off > 0; off >>= 1) lmax = fmaxf(lmax, __shfl_xor(lmax, off));
  if ((tid & 31) == 0) red[tid >> 5] = lmax;
  __syncthreads();
  if (tid == 0) {
    float mm = red[0];
    for (int w = 1; w < TPB / 32; ++w) mm = fmaxf(mm, red[w]);
    s_m = mm * invT;                                      // max of scaled logits
    s_prefix = 0;
    s_remaining = TOPK;
    radix_resolve(hist, &s_prefix, &s_remaining);
  }
  __syncthreads();
  const float m = s_m;
  uint32_t pfx = s_prefix;

  // ---------------- Pass B: sum(exp) + radix round 1 (bits [23:16]) -------
  hist[tid] = 0;
  __syncthreads();
  float lsum = 0.0f;
  for (int i = tid * 4; i < V4; i += TPB * 4) {
    __builtin_prefetch(lrow + i + TPB * 16, 0, 3);
    float4 x = *reinterpret_cast<const float4*>(lrow + i);
    lsum += expf(__builtin_fmaf(x.x, invT, -m));
    lsum += expf(__builtin_fmaf(x.y, invT, -m));
    lsum += expf(__builtin_fmaf(x.z, invT, -m));
    lsum += expf(__builtin_fmaf(x.w, invT, -m));
    uint32_t k0 = f2ord(x.x), k1 = f2ord(x.y), k2 = f2ord(x.z), k3 = f2ord(x.w);
    if ((k0 >> 24) == pfx) atomicAdd(&hist[(k0 >> 16) & 0xFFu], 1u);
    if ((k1 >> 24) == pfx) atomicAdd(&hist[(k1 >> 16) & 0xFFu], 1u);
    if ((k2 >> 24) == pfx) atomicAdd(&hist[(k2 >> 16) & 0xFFu], 1u);
    if ((k3 >> 24) == pfx) atomicAdd(&hist[(k3 >> 16) & 0xFFu], 1u);
  }
  for (int i = V4 + tid; i < V; i += TPB) {
    float x = lrow[i];
    lsum += expf(__builtin_fmaf(x, invT, -m));
    uint32_t k = f2ord(x);
    if ((k >> 24) == pfx) atomicAdd(&hist[(k >> 16) & 0xFFu], 1u);
  }
  for (int off = 16; off > 0; off >>= 1) lsum += __shfl_xor(lsum, off);
  if ((tid & 31) == 0) red[tid >> 5] = lsum;
  __syncthreads();
  if (tid == 0) {
    float ss = red[0];
    for (int w = 1; w < TPB / 32; ++w) ss += red[w];
    s_lse = logf(ss);
    radix_resolve(hist, &s_prefix, &s_remaining);
  }
  __syncthreads();
  const float lse = s_lse;
  pfx = s_prefix;

  // ---------------- Pass C: write logprobs + radix round 2 (bits [15:8]) --
  hist[tid] = 0;
  __syncthreads();
  for (int i = tid * 4; i < V4; i += TPB * 4) {
    float4 x = *reinterpret_cast<const float4*>(lrow + i);
    float4 o;
    o.x = __builtin_fmaf(x.x, invT, -m) - lse;
    o.y = __builtin_fmaf(x.y, invT, -m) - lse;
    o.z = __builtin_fmaf(x.z, invT, -m) - lse;
    o.w = __builtin_fmaf(x.w, invT, -m) - lse;
    *reinterpret_cast<float4*>(lpout + i) = o;
    uint32_t k0 = f2ord(x.x), k1 = f2ord(x.y), k2 = f2ord(x.z), k3 = f2ord(x.w);
    if ((k0 >> 16) == pfx) atomicAdd(&hist[(k0 >> 8) & 0xFFu], 1u);
    if ((k1 >> 16) == pfx) atomicAdd(&hist[(k1 >> 8) & 0xFFu], 1u);
    if ((k2 >> 16) == pfx) atomicAdd(&hist[(k2 >> 8) & 0xFFu], 1u);
    if ((k3 >> 16) == pfx) atomicAdd(&hist[(k3 >> 8) & 0xFFu], 1u);
  }
  for (int i = V4 + tid; i < V; i += TPB) {
    float x = lrow[i];
    lpout[i] = __builtin_fmaf(x, invT, -m) - lse;
    uint32_t k = f2ord(x);
    if ((k >> 16) == pfx) atomicAdd(&hist[(k >> 8) & 0xFFu], 1u);
  }
  __syncthreads();
  if (tid == 0) radix_resolve(hist, &s_prefix, &s_remaining);
  __syncthreads();
  pfx = s_prefix;

  // ---------------- Pass D: radix round 3 (bits [7:0]) via TDM staging ----
  hist[tid] = 0;
  __syncthreads();
#if __has_builtin(__builtin_amdgcn_tensor_load_to_lds) && __has_builtin(__builtin_amdgcn_s_wait_tensorcnt)
  {
    const uint32_t stage_off = (uint32_t)(uintptr_t)&stage[0];  // LDS byte offset
    for (int off = 0; off < V; off += CHUNK) {
      const int n = (V - off < CHUNK) ? (V - off) : CHUNK;
      if (tid < 32) {                                    // wave 0 drives the DMA
        tdm_load_row_chunk(stage_off, lrow + off, (uint32_t)n);
        __builtin_amdgcn_s_wait_tensorcnt(0);
      }
      __syncthreads();                                   // LDS chunk visible to all waves
      for (int j = tid; j < n; j += TPB) {
        uint32_t k = f2ord(stage[j]);
        if ((k >> 8) == pfx) atomicAdd(&hist[k & 0xFFu], 1u);
      }
      __syncthreads();                                   // chunk consumed, buffer reusable
    }
  }
#else
  for (int i = tid; i < V; i += TPB) {
    uint32_t k = f2ord(lrow[i]);
    if ((k >> 8) == pfx) atomicAdd(&hist[k & 0xFFu], 1u);
  }
  __syncthreads();
#endif
  if (tid == 0) {
    radix_resolve(hist, &s_prefix, &s_remaining);
    s_thr = s_prefix;                                    // full key of the 64th-largest
    s_ccount = 0;
  }
  __syncthreads();
  const uint32_t thr = s_thr;

  // ---------------- Pass E: gather candidates with key >= thr -------------
  cand[tid] = ~0ULL;
  __syncthreads();
  for (int i = tid * 4; i < V4; i += TPB * 4) {
    float4 x = *reinterpret_cast<const float4*>(lrow + i);
    uint32_t k0 = f2ord(x.x), k1 = f2ord(x.y), k2 = f2ord(x.z), k3 = f2ord(x.w);
    if (k0 >= thr) { uint32_t p = atomicAdd(&s_ccount, 1u); if (p < CAP) cand[p] = ((unsigned long long)(~k0) << 32) | (uint32_t)(i + 0); }
    if (k1 >= thr) { uint32_t p = atomicAdd(&s_ccount, 1u); if (p < CAP) cand[p] = ((unsigned long long)(~k1) << 32) | (uint32_t)(i + 1); }
    if (k2 >= thr) { uint32_t p = atomicAdd(&s_ccount, 1u); if (p < CAP) cand[p] = ((unsigned long long)(~k2) << 32) | (uint32_t)(i + 2); }
    if (k3 >= thr) { uint32_t p = atomicAdd(&s_ccount, 1u); if (p < CAP) cand[p] = ((unsigned long long)(~k3) << 32) | (uint32_t)(i + 3); }
  }
  for (int i = V4 + tid; i < V; i += TPB) {
    uint32_t k = f2ord(lrow[i]);
    if (k >= thr) { uint32_t p = atomicAdd(&s_ccount, 1u); if (p < CAP) cand[p] = ((unsigned long long)(~k) << 32) | (uint32_t)i; }
  }
  __syncthreads();

  // ---------------- Bitonic sort: value desc, index asc -------------------
  for (unsigned k2s = 2; k2s <= CAP; k2s <<= 1) {
    for (unsigned j = k2s >> 1; j > 0; j >>= 1) {
      unsigned i = tid, ixj = i ^ j;
      if (ixj > i) {
        bool asc = ((i & k2s) == 0);
        unsigned long long a = cand[i], b = cand[ixj];
        if ((a > b) == asc) { cand[i] = b; cand[ixj] = a; }
      }
      __syncthreads();
    }
  }

  // ---------------- Sequential top-k / top-p / min-p / sample -------------
  if (tid == 0) {
    const int   kk = topks[row];
    const float tp = topps[row];
    const float mp = minps[row];
    const float uu = us[row];

    float cumAll = 0.0f, total = 0.0f, thrmp = 0.0f;
    for (int r = 0; r < TOPK; ++r) {
      unsigned long long c = cand[r];
      float p = 0.0f;
      if (c != ~0ULL) {
        uint32_t key = ~(uint32_t)(c >> 32);
        p = expf(__builtin_fmaf(ord2f(key), invT, -m) - lse);
      }
      cumAll += p;
      bool keep = (r < kk) && (cumAll - p <= tp);
      float pf = keep ? p : 0.0f;
      if (r == 0) thrmp = pf * mp;                        // p0 * min_p (rank 0 always kept)
      if (pf < thrmp) pf = 0.0f;
      red[r] = pf;
      total += pf;
    }
    const float target = uu * total;
    float cdf = 0.0f;
    int sel = TOPK - 1;
    for (int r = 0; r < TOPK; ++r) {
      cdf += red[r];
      if (cdf >= target) { sel = r; break; }
    }
    idout[row] = (int)(uint32_t)cand[sel];                // low 32 bits = token index
  }
}

extern "C" void kernel_launch(void* const* d_in, const int* in_sizes, int n_in,
                              void* d_out, int out_size, void* d_ws, size_t ws_size,
                              hipStream_t stream) {
  const float* logits = (const float*)d_in[0];
  const float* temps  = (const float*)d_in[1];
  const int*   topks  = (const int*)  d_in[2];
  const float* topps  = (const float*)d_in[3];
  const float* minps  = (const float*)d_in[4];
  const float* us     = (const float*)d_in[5];

  const int B = in_sizes[1];
  const int V = in_sizes[0] / B;

  sampler_topk_kernel<<<B, TPB, 0, stream>>>(
      logits, temps, topks, topps, minps, us, (float*)d_out, B, V);
}